// ChamferLoss_21930103013751
// MI455X (gfx1250) — compile-verified
//
#include <hip/hip_runtime.h>
#include <math.h>
#include <stdint.h>

// Chamfer loss, B=8, D=3, N=4096.
// ws layout (floats): [0, B*N)          = colMin (min over predict rows, per gt point)
//                     [B*N, 2*B*N)      = rowMin (min over gt points, per predict row)

#define BATCH 8
#define NPTS 4096
#define ROWS_PER_BLOCK 128   // 8 waves x 16 rows
#define CHUNK 512            // gt points staged in LDS per iteration
#define WAVES 8

typedef __attribute__((ext_vector_type(2))) float v2f;
typedef __attribute__((ext_vector_type(8))) float v8f;

__device__ __forceinline__ void global_atomic_min_f32(float* p, float v) {
  asm volatile("global_atomic_min_num_f32 %0, %1, off scope:SCOPE_DEV"
               :: "v"((uint64_t)(uintptr_t)p), "v"(v)
               : "memory");
}

__global__ __launch_bounds__(256) void chamfer_init(float* __restrict__ ws) {
  int idx = blockIdx.x * blockDim.x + threadIdx.x;
  if (idx < 2 * BATCH * NPTS) ws[idx] = INFINITY;
}

__global__ __launch_bounds__(256) void chamfer_main(const float* __restrict__ P,
                                                    const float* __restrict__ G,
                                                    float* __restrict__ ws) {
  __shared__ float4 sG[CHUNK];                     // (-2gx, -2gy, -2gz, |g|^2)   8 KB
  __shared__ float  sColHalf[2 * WAVES * CHUNK];   // per-(wave,half) col minima 32 KB

  const int tid   = threadIdx.x;
  const int lane  = tid & 31;
  const int wave  = tid >> 5;
  const int b     = blockIdx.x >> 5;   // 32 row-blocks per batch
  const int iBlk  = blockIdx.x & 31;
  const int iBase = iBlk * ROWS_PER_BLOCK + wave * 16;

  const float* Px = P + (size_t)b * 3 * NPTS;
  const float* Py = Px + NPTS;
  const float* Pz = Py + NPTS;
  const float* Gx = G + (size_t)b * 3 * NPTS;
  const float* Gy = Gx + NPTS;
  const float* Gz = Gy + NPTS;

  float* colMin = ws + (size_t)b * NPTS;
  float* rowMin = ws + (size_t)BATCH * NPTS + (size_t)b * NPTS;

  // ---- A fragment: 16x4 f32.  Lanes 0-15 hold (K0,K1)=(x,y) for row M=lane;
  //      lanes 16-31 hold (K2,K3)=(z,1) for row M=lane-16.
  const int  m  = lane & 15;
  const bool hi = lane >= 16;
  float ax = Px[iBase + m], ay = Py[iBase + m], az = Pz[iBase + m];
  v2f afrag;
  afrag.x = hi ? az : ax;
  afrag.y = hi ? 1.0f : ay;

  // ---- per-C-row addend |p_i|^2 (C layout: VGPR r -> row r (lanes 0-15) / r+8 (lanes 16-31))
  float addp[8];
  const int rBase = hi ? 8 : 0;
#pragma unroll
  for (int r = 0; r < 8; ++r) {
    int i2 = iBase + rBase + r;
    float x = Px[i2], y = Py[i2], z = Pz[i2];
    addp[r] = x * x + y * y + z * z;
  }

  float rm[8];
#pragma unroll
  for (int r = 0; r < 8; ++r) rm[r] = INFINITY;

  // each (wave, lane-half) owns its own column-min slice -> no divergence, no bpermute
  float* myCol = &sColHalf[(wave * 2 + (hi ? 1 : 0)) * CHUNK];

  for (int chunk = 0; chunk < NPTS / CHUNK; ++chunk) {
    const int jc0 = chunk * CHUNK;
    __syncthreads();  // previous chunk's sG / sColHalf fully consumed

    // stage gt chunk into LDS as (-2g, |g|^2)
#pragma unroll
    for (int k = 0; k < CHUNK / 256; ++k) {
      int jc = tid + k * 256;
      int j  = jc0 + jc;
      float gx = Gx[j], gy = Gy[j], gz = Gz[j];
      sG[jc] = make_float4(-2.0f * gx, -2.0f * gy, -2.0f * gz,
                           gx * gx + gy * gy + gz * gz);
    }
    if (chunk + 1 < NPTS / CHUNK) {  // prefetch next chunk (global_prefetch_b8)
      __builtin_prefetch(Gx + jc0 + CHUNK + tid * 2, 0, 0);
      __builtin_prefetch(Gy + jc0 + CHUNK + tid * 2, 0, 0);
      __builtin_prefetch(Gz + jc0 + CHUNK + tid * 2, 0, 0);
    }
    __syncthreads();

    // each wave sweeps all 32 column tiles of this chunk against its 16 rows
    for (int tile = 0; tile < CHUNK / 16; ++tile) {
      int jc = tile * 16 + m;
      float4 g = sG[jc];  // ds_load_b128

      // B fragment: 4x16 f32. Lanes 0-15 hold K rows (0,1); lanes 16-31 rows (2,3).
      v2f bfrag;
      bfrag.x = hi ? g.z : g.x;
      bfrag.y = hi ? g.w : g.y;

      v8f c = {};
      // E = A x B:  E_ij = |g_j|^2 - 2 <p_i, g_j>
      v8f e = __builtin_amdgcn_wmma_f32_16x16x4_f32(
          false, afrag, false, bfrag, (short)0, c, false, false);

      float colm = INFINITY;
#pragma unroll
      for (int r = 0; r < 8; ++r) {
        float d2 = e[r] + addp[r];        // full squared distance
        rm[r]    = fminf(rm[r], d2);      // min over j (row term)
        colm     = fminf(colm, d2);       // min over this lane-half's 8 rows
      }
      myCol[jc] = colm;  // all 32 lanes store, unique addresses, no exec mask
    }
    __syncthreads();

    // reduce column minima across the 16 (wave,half) slices, push with float atomic-min
#pragma unroll
    for (int k = 0; k < CHUNK / 256; ++k) {
      int jc = tid + k * 256;
      float mn = sColHalf[jc];
#pragma unroll
      for (int h = 1; h < 2 * WAVES; ++h) mn = fminf(mn, sColHalf[h * CHUNK + jc]);
      global_atomic_min_f32(&colMin[jc0 + jc], mn);
    }
  }

  // ---- row minima: reduce each 16-lane half across the N (=lane%16) dimension
#pragma unroll
  for (int r = 0; r < 8; ++r) {
    float v = rm[r];
    v = fminf(v, __shfl_xor(v, 1, 32));
    v = fminf(v, __shfl_xor(v, 2, 32));
    v = fminf(v, __shfl_xor(v, 4, 32));
    v = fminf(v, __shfl_xor(v, 8, 32));
    if (lane == 0)  rowMin[iBase + r]     = v;  // rows 0..7 of this wave tile
    if (lane == 16) rowMin[iBase + 8 + r] = v;  // rows 8..15
  }
}

__global__ __launch_bounds__(256) void chamfer_final(const float* __restrict__ ws,
                                                     float* __restrict__ out) {
  __shared__ float red[256];
  float s = 0.0f;
  for (int idx = threadIdx.x; idx < 2 * BATCH * NPTS; idx += 256)
    s += sqrtf(fmaxf(ws[idx], 1e-12f));   // sqrt deferred to the 65536 minima
  red[threadIdx.x] = s;
  __syncthreads();
  for (int off = 128; off > 0; off >>= 1) {
    if (threadIdx.x < off) red[threadIdx.x] += red[threadIdx.x + off];
    __syncthreads();
  }
  if (threadIdx.x == 0) out[0] = red[0] / (float)(BATCH * NPTS);
}

extern "C" void kernel_launch(void* const* d_in, const int* in_sizes, int n_in,
                              void* d_out, int out_size, void* d_ws, size_t ws_size,
                              hipStream_t stream) {
  const float* P = (const float*)d_in[0];  // predict_pc [8,3,4096]
  const float* G = (const float*)d_in[1];  // gt_pc      [8,3,4096]
  float* ws = (float*)d_ws;                // 2*B*N floats = 256 KB

  chamfer_init<<<(2 * BATCH * NPTS + 255) / 256, 256, 0, stream>>>(ws);
  chamfer_main<<<BATCH * (NPTS / ROWS_PER_BLOCK), 256, 0, stream>>>(P, G, ws);
  chamfer_final<<<1, 256, 0, stream>>>(ws, (float*)d_out);
}